// LayerPPoly_9354438770804
// MI455X (gfx1250) — compile-verified
//
#include <hip/hip_runtime.h>

// Problem constants (from reference): K=4, M=128, N=32, D=7, NQ=8388608
#define K_ 4
#define M_ 128
#define N_ 32
#define D_ 7
#define ROW_ (K_ * D_)        // 28 floats of coefficients per segment
#define TAB_ (M_ * ROW_)      // 3584 floats = 14336 B staged in LDS
#define THREADS 256
#define QPT 16                // queries per thread
#define QPB (THREADS * QPT)   // 4096 queries per block

#if __has_builtin(__builtin_amdgcn_global_load_async_to_lds_b32)
#define HAVE_ASYNC_LDS 1
#else
#define HAVE_ASYNC_LDS 0
#endif

// Builtin signature (from hipcc diagnostic): int* in addrspace(1)/(3).
typedef __attribute__((address_space(1))) int gint_t;
typedef __attribute__((address_space(3))) int lint_t;

// Evaluate one query: branchless segment search + Horner over staged table.
__device__ __forceinline__ void eval_query(float x,
                                           const float* __restrict__ s_xb,
                                           const float* __restrict__ s_tab,
                                           float* __restrict__ y /*[7]*/) {
  // xb = linspace(0,1,129) has step 2^-7 (exact), so floor(x*128) is already
  // the searchsorted segment; a single +-1 branchless fix-up covers any ulp
  // deviation. Clamp to [0, 127].
  int seg = (int)(x * (float)M_);
  seg = seg < 0 ? 0 : (seg > M_ - 1 ? M_ - 1 : seg);
  const bool dec = (seg > 0) && (x < s_xb[seg]);
  const bool inc = (seg < M_ - 1) && (x >= s_xb[seg + 1]);
  seg += (int)inc - (int)dec;
  const float t = x - s_xb[seg];

  // 28 contiguous, 16B-aligned floats: 7x ds_load_b128
  const float4* row = (const float4*)(s_tab + seg * ROW_);
  __align__(16) float f[ROW_];
#pragma unroll
  for (int w = 0; w < 7; ++w) ((float4*)f)[w] = row[w];

#pragma unroll
  for (int d = 0; d < D_; ++d) {
    float acc = f[d];                       // ck[0]
    acc = fmaf(acc, t, f[1 * D_ + d]);      // *t + ck[1]
    acc = fmaf(acc, t, f[2 * D_ + d]);      // *t + ck[2]
    acc = fmaf(acc, t, f[3 * D_ + d]);      // *t + ck[3]
    y[d] = acc;
  }
}

__global__ __launch_bounds__(THREADS) void ppoly_kernel(
    const float* __restrict__ xq,   // (NQ,)
    const float* __restrict__ c,    // (K, M, N, N, D)
    const float* __restrict__ xb,   // (M+1,)
    const int* __restrict__ ip,     // scalar i
    const int* __restrict__ jp,     // scalar j
    float* __restrict__ out,        // (NQ, D)
    int nq) {
  __shared__ __align__(16) float s_tab[TAB_];
  __shared__ float s_xb[M_ + 1];

  const int tid = threadIdx.x;
  const int ii = *ip;
  const int jj = *jp;
  const size_t cbase = (size_t)ii * (N_ * D_) + (size_t)jj * D_;

  // Stage breakpoints.
  if (tid <= M_) s_xb[tid] = xb[tid];

  // Stage the live 14KB slice c[k, m, i, j, :] into LDS as
  // s_tab[m*28 + k*7 + d] via the CDNA5 async global->LDS data mover.
  // 512 rows of 7 floats; row id r = m*4 + k -> shifts/masks only.
#pragma unroll
  for (int rr = 0; rr < (TAB_ / D_) / THREADS; ++rr) {   // 2 rows per thread
    const int r = tid + rr * THREADS;                    // 0..511
    const int m = r >> 2;
    const int k = r & 3;
    const float* gsrc = c + (size_t)k * (M_ * N_ * N_ * D_) +
                        (size_t)m * (N_ * N_ * D_) + cbase;
    const int loff = m * ROW_ + k * D_;
#pragma unroll
    for (int d = 0; d < D_; ++d) {
#if HAVE_ASYNC_LDS
      __builtin_amdgcn_global_load_async_to_lds_b32(
          (gint_t*)(gsrc + d), (lint_t*)(s_tab + loff + d), 0, 0);
#else
      s_tab[loff + d] = gsrc[d];
#endif
    }
  }
  __builtin_amdgcn_s_wait_asynccnt(0);  // ASYNCcnt == 0: LDS tile resident
  __syncthreads();

  const int qbase = (int)blockIdx.x * QPB;

#pragma unroll
  for (int it = 0; it < QPT / 4; ++it) {
    const int v = it * THREADS + tid;   // float4 slot within block
    const int q0 = qbase + v * 4;       // first query of this slot

    if (q0 + 4 <= nq) {
      // Coalesced 16B load of 4 queries.
      float4 q4 = *(const float4*)(xq + q0);
      if (it + 1 < QPT / 4)
        __builtin_prefetch(xq + q0 + THREADS * 4, 0, 0);  // global_prefetch_b8

      __align__(16) float res[4 * D_];  // 28 contiguous output floats
      float qs[4] = {q4.x, q4.y, q4.z, q4.w};
#pragma unroll
      for (int e = 0; e < 4; ++e)
        eval_query(qs[e], s_xb, s_tab, res + e * D_);

      // q0 % 4 == 0 -> byte offset q0*28 is 16B aligned: 7x global_store_b128
      float4* o4 = (float4*)(out + (size_t)q0 * D_);
#pragma unroll
      for (int w = 0; w < 7; ++w) o4[w] = ((const float4*)res)[w];
    } else if (q0 < nq) {
      // Tail (not taken for NQ = 2^23, kept for generality).
      for (int e = 0; e < 4 && (q0 + e) < nq; ++e) {
        float y[D_];
        eval_query(xq[q0 + e], s_xb, s_tab, y);
        for (int d = 0; d < D_; ++d) out[(size_t)(q0 + e) * D_ + d] = y[d];
      }
    }
  }
}

extern "C" void kernel_launch(void* const* d_in, const int* in_sizes, int n_in,
                              void* d_out, int out_size, void* d_ws,
                              size_t ws_size, hipStream_t stream) {
  const float* xq = (const float*)d_in[0];
  const float* c  = (const float*)d_in[1];
  const float* xb = (const float*)d_in[2];
  const int* ip   = (const int*)d_in[3];
  const int* jp   = (const int*)d_in[4];
  float* out      = (float*)d_out;

  const int nq = in_sizes[0];
  const int blocks = (nq + QPB - 1) / QPB;  // 2048 for NQ = 8388608

  ppoly_kernel<<<dim3(blocks), dim3(THREADS), 0, stream>>>(
      xq, c, xb, ip, jp, out, nq);
}